// DetectionCriterion_29171417874567
// MI455X (gfx1250) — compile-verified
//
#include <hip/hip_runtime.h>
#include <math.h>

// DetectionCriterion loss: fully-reduced scalar over (B=2e6, A=3) anchors.
// output: [B,A,7] f32  (obj, x, y, z, cls0, cls1, cls2)
// target: [B,A,5] f32  (obj, x, y, z, cls_idx)
// loss = 2*class_bce + 4*xywh_sq + obj_bce  (summed over everything)
//
// Bandwidth-bound: 288 MB streamed once -> ~12.4 us floor at 23.3 TB/s.
// 4 anchors/thread => 16B-aligned b128 nontemporal loads (112B / 80B rows).
// BCE with binary targets needs ONE log per value: log(t ? p : 1-p).
// Wave reduction via v_wmma_f32_16x16x4_f32 (A=partials, B=ones) + SWAPX16.

typedef float v2f __attribute__((ext_vector_type(2)));
typedef float v4f __attribute__((ext_vector_type(4)));
typedef float v8f __attribute__((ext_vector_type(8)));

#define NCLS 3
#define BLK 256
#define GRID_BLOCKS 2048

// -max(log(sel), -100): single hardware log per BCE value.
// v_log_f32 * ln2; p in (1e-4, 1-1e-4) so the -100 clamp never binds but is kept.
__device__ __forceinline__ float bce_one_log(float p, bool t_one) {
    float arg = t_one ? p : (1.0f - p);
    return -fmaxf(__logf(arg), -100.0f);
}

// Per-anchor loss; o[0..6], t[0..4] must be register-resident (constant-indexed).
__device__ __forceinline__ float anchor_loss(const float* o, const float* t) {
    float obj_loss = bce_one_log(o[0], t[0] == 1.0f);
    float dx = o[1] - t[1];
    float dy = o[2] - t[2];
    float dz = o[3] - t[3];
    float xywh = dx * dx + dy * dy + dz * dz;
    int cls = (int)t[4];
    float class_loss = 0.0f;
    #pragma unroll
    for (int c = 0; c < NCLS; ++c)
        class_loss += bce_one_log(o[4 + c], c == cls);
    return 2.0f * class_loss + 4.0f * xywh + obj_loss;
}

// Swap lanes 0-15 <-> 16-31: ds_swizzle group-of-32, xor_mask=0x10, and_mask=0x1f.
__device__ __forceinline__ float swap_half(float x) {
    return __int_as_float(__builtin_amdgcn_ds_swizzle(__float_as_int(x), 0x401F));
}

// Exact wave32 sum on the matrix pipe:
// A (16x4 f32): VGPR0 = acc (A[m][0]=acc_m, A[m][2]=acc_{m+16}), VGPR1 = 0.
// B (4x16) = ones  =>  D[m][n] = acc_m + acc_{m+16}.
// Lane sums its 8 D VGPRs (half-set sum), SWAPX16 merges the two halves.
__device__ __forceinline__ float wave_reduce_sum(float acc) {
    v2f a; a.x = acc;  a.y = 0.0f;
    v2f b; b.x = 1.0f; b.y = 1.0f;
    v8f c = {};
    v8f d = __builtin_amdgcn_wmma_f32_16x16x4_f32(
        /*neg_a=*/false, a, /*neg_b=*/false, b,
        /*c_mod=*/(short)0, c, /*reuse_a=*/false, /*reuse_b=*/false);
    float t = ((d[0] + d[1]) + (d[2] + d[3])) + ((d[4] + d[5]) + (d[6] + d[7]));
    return t + swap_half(t);   // full wave sum in every lane
}

__device__ __forceinline__ float block_reduce_sum(float acc) {
    __shared__ float wsum[BLK / 32];
    float w = wave_reduce_sum(acc);
    if ((threadIdx.x & 31) == 0) wsum[threadIdx.x >> 5] = w;
    __syncthreads();
    float s = 0.0f;
    if (threadIdx.x == 0) {
        #pragma unroll
        for (int i = 0; i < BLK / 32; ++i) s += wsum[i];  // fixed order: deterministic
    }
    return s;  // valid on thread 0 only
}

__global__ void det_loss_partial(const float* __restrict__ out_p,
                                 const float* __restrict__ tgt_p,
                                 float* __restrict__ partial,
                                 int n_groups, int n_anchors) {
    int tid    = blockIdx.x * blockDim.x + threadIdx.x;
    int stride = gridDim.x * blockDim.x;
    float acc = 0.0f;

    for (int g = tid; g < n_groups; g += stride) {
        float o[28];  // 4 anchors x 7
        float t[20];  // 4 anchors x 5
        const v4f* op = (const v4f*)(out_p + (size_t)g * 28);  // 112B, 16B-aligned
        const v4f* tp = (const v4f*)(tgt_p + (size_t)g * 20);  //  80B, 16B-aligned
        #pragma unroll
        for (int j = 0; j < 7; ++j) ((v4f*)o)[j] = __builtin_nontemporal_load(op + j);
        #pragma unroll
        for (int j = 0; j < 5; ++j) ((v4f*)t)[j] = __builtin_nontemporal_load(tp + j);
        #pragma unroll
        for (int k = 0; k < 4; ++k)
            acc += anchor_loss(o + k * 7, t + k * 5);
    }

    // Deterministic scalar tail (N % 4 anchors), handled by one thread.
    if (tid == 0) {
        for (int i = n_groups * 4; i < n_anchors; ++i) {
            float o[7], t[5];
            #pragma unroll
            for (int j = 0; j < 7; ++j) o[j] = out_p[(size_t)i * 7 + j];
            #pragma unroll
            for (int j = 0; j < 5; ++j) t[j] = tgt_p[(size_t)i * 5 + j];
            acc += anchor_loss(o, t);
        }
    }

    float s = block_reduce_sum(acc);   // EXEC full here (reconverged)
    if (threadIdx.x == 0) partial[blockIdx.x] = s;
}

__global__ void det_loss_final(const float* __restrict__ partial,
                               float* __restrict__ out,
                               int n_partial) {
    float acc = 0.0f;
    for (int i = threadIdx.x; i < n_partial; i += BLK)  // fixed order per thread
        acc += partial[i];
    float s = block_reduce_sum(acc);
    if (threadIdx.x == 0) out[0] = s;
}

extern "C" void kernel_launch(void* const* d_in, const int* in_sizes, int n_in,
                              void* d_out, int out_size, void* d_ws, size_t ws_size,
                              hipStream_t stream) {
    const float* output = (const float*)d_in[0];  // [B,A,7]
    const float* target = (const float*)d_in[1];  // [B,A,5]
    float* out  = (float*)d_out;
    float* part = (float*)d_ws;

    int n_anchors = in_sizes[0] / 7;   // B*A = 6,000,000
    int n_groups  = n_anchors / 4;     // 1,500,000 (exact for this shape)

    int grid = GRID_BLOCKS;
    if (ws_size < (size_t)grid * sizeof(float))
        grid = (int)(ws_size / sizeof(float));
    if (grid < 1) grid = 1;

    det_loss_partial<<<grid, BLK, 0, stream>>>(output, target, part, n_groups, n_anchors);
    det_loss_final<<<1, BLK, 0, stream>>>(part, out, grid);
}